// GAAEncoder_81655918232324
// MI455X (gfx1250) — compile-verified
//
#include <hip/hip_runtime.h>

typedef __attribute__((ext_vector_type(2))) float v2f;
typedef __attribute__((ext_vector_type(8))) float v8f;

#define NEG_SLOPE 0.2f

// ---------- helpers ----------

// Monotone float -> uint encoding for atomicMax-based segment max.
__device__ __forceinline__ unsigned enc_f32(float f) {
  unsigned u = __float_as_uint(f);
  return (u & 0x80000000u) ? ~u : (u | 0x80000000u);
}
__device__ __forceinline__ float dec_f32(unsigned k) {
  if (k == 0u) return 0.0f;  // untouched segment -> reference uses 0.0
  unsigned u = (k & 0x80000000u) ? (k & 0x7FFFFFFFu) : ~k;
  return __uint_as_float(u);
}

__device__ __forceinline__ void atomAddF(float* p, float v) {
  unsafeAtomicAdd(p, v);  // lowers to global_atomic_add_f32
}

__device__ __forceinline__ v8f wmma4(v2f a, v2f b, v8f c) {
  return __builtin_amdgcn_wmma_f32_16x16x4_f32(false, a, false, b, (short)0, c,
                                               false, false);
}

// ---------- kernels ----------

__global__ void k_zero(float* __restrict__ p, long n) {
  long i = (long)blockIdx.x * blockDim.x + threadIdx.x;
  long stride = (long)gridDim.x * blockDim.x;
  for (; i < n; i += stride) p[i] = 0.0f;
}

// C[M,N] = A[M,K] @ B[K,N], fp32 WMMA 16x16x4. One wave -> 16x64 C tile.
// A frag (ISA 7.12.2, 32-bit A 16x4): lanes0-15 M=lane {V0=K0,V1=K1},
// lanes16-31 M=lane-16 {V0=K2,V1=K3}. B/C: row striped across lanes per VGPR.
// Software-pipelined: next k-step fragments load while current WMMAs execute.
__global__ void gemm_wmma(const float* __restrict__ A, const float* __restrict__ B,
                          float* __restrict__ C, int M, int K, int N) {
  int wave = (blockIdx.x * blockDim.x + threadIdx.x) >> 5;
  int lane = threadIdx.x & 31;
  int tilesM = (M + 15) >> 4;
  int tilesN = N >> 6;  // N is a multiple of 64 here (256 or 128)
  if (wave >= tilesM * tilesN) return;
  int tm = wave % tilesM;
  int tn = wave / tilesM;
  int half = lane >> 4;   // selects K-pair for A, V-row pair for B/C
  int l15  = lane & 15;

  int row  = tm * 16 + l15;
  int arow = row < M ? row : (M - 1);   // clamp: keep EXEC all-1s for WMMA
  int n0 = tn * 64;

  // running pointers (avoid per-iteration 64-bit index mults)
  const float* Ap  = A + (size_t)arow * K + half * 2;          // advance by 4
  const float* Bp0 = B + (size_t)(half * 2) * N + n0 + l15;    // advance by 4*N
  const float* Bp1 = Bp0 + N;
  const int bstep = 4 * N;

  v8f c0 = {}, c1 = {}, c2 = {}, c3 = {};

  // prologue: fragments for kb = 0
  v2f a = *(const v2f*)Ap;
  v2f b0, b1, b2, b3;
  b0.x = Bp0[0];  b0.y = Bp1[0];
  b1.x = Bp0[16]; b1.y = Bp1[16];
  b2.x = Bp0[32]; b2.y = Bp1[32];
  b3.x = Bp0[48]; b3.y = Bp1[48];

  for (int kb = 4; kb < K; kb += 4) {
    Ap += 4; Bp0 += bstep; Bp1 += bstep;
    // issue next k-step loads before consuming current fragments
    v2f an = *(const v2f*)Ap;
    v2f n0f, n1f, n2f, n3f;
    n0f.x = Bp0[0];  n0f.y = Bp1[0];
    n1f.x = Bp0[16]; n1f.y = Bp1[16];
    n2f.x = Bp0[32]; n2f.y = Bp1[32];
    n3f.x = Bp0[48]; n3f.y = Bp1[48];

    c0 = wmma4(a, b0, c0);
    c1 = wmma4(a, b1, c1);
    c2 = wmma4(a, b2, c2);
    c3 = wmma4(a, b3, c3);

    a = an; b0 = n0f; b1 = n1f; b2 = n2f; b3 = n3f;
  }
  c0 = wmma4(a, b0, c0);
  c1 = wmma4(a, b1, c1);
  c2 = wmma4(a, b2, c2);
  c3 = wmma4(a, b3, c3);

  union V8 { v8f v; float f[8]; } u0, u1, u2, u3;
  u0.v = c0; u1.v = c1; u2.v = c2; u3.v = c3;
  for (int v = 0; v < 8; ++v) {
    int r = tm * 16 + v + half * 8;  // C VGPR v: M = v (lanes0-15) / v+8 (16-31)
    if (r < M) {
      float* Cr = C + (size_t)r * N + n0 + l15;
      Cr[0]  = u0.f[v];
      Cr[16] = u1.f[v];
      Cr[32] = u2.f[v];
      Cr[48] = u3.f[v];
    }
  }
}

// Per-node attention logits: as[n]=h[n]·a_src, ad[n]=h[n]·a_dst. One wave/node.
__global__ void alpha_dots(const float* __restrict__ H, const float* __restrict__ asrc,
                           const float* __restrict__ adst, float* __restrict__ outs,
                           float* __restrict__ outd, int Nn, int F) {
  int wave = (blockIdx.x * blockDim.x + threadIdx.x) >> 5;
  int lane = threadIdx.x & 31;
  if (wave >= Nn) return;
  const float* h = H + (size_t)wave * F;
  float s0 = 0.f, s1 = 0.f;
  for (int f = lane; f < F; f += 32) {
    float v = h[f];
    s0 += v * asrc[f];
    s1 += v * adst[f];
  }
  for (int m = 16; m; m >>= 1) {
    s0 += __shfl_xor(s0, m, 32);
    s1 += __shfl_xor(s1, m, 32);
  }
  if (lane == 0) { outs[wave] = s0; outd[wave] = s1; }
}

// Pass 1: segment max of LeakyReLU(as[src]+ad[dst]) over dst.
__global__ void edge_max(const int* __restrict__ src, const int* __restrict__ dst,
                         const float* __restrict__ as, const float* __restrict__ ad,
                         unsigned* __restrict__ emax, int E) {
  int i = blockIdx.x * blockDim.x + threadIdx.x;
  if (i >= E) return;
  int s = src[i], d = dst[i];
  float e = as[s] + ad[d];
  e = e > 0.f ? e : NEG_SLOPE * e;
  atomicMax(&emax[d], enc_f32(e));
}

// Pass 2: exponent per edge (cached) + segment sum of exponents.
__global__ void edge_exp(const int* __restrict__ src, const int* __restrict__ dst,
                         const float* __restrict__ as, const float* __restrict__ ad,
                         const unsigned* __restrict__ emax, float* __restrict__ denom,
                         float* __restrict__ exb, int E) {
  int i = blockIdx.x * blockDim.x + threadIdx.x;
  if (i >= E) return;
  int s = src[i], d = dst[i];
  float e = as[s] + ad[d];
  e = e > 0.f ? e : NEG_SLOPE * e;
  float ex = __expf(e - dec_f32(emax[d]));
  exb[i] = ex;
  atomAddF(&denom[d], ex);
}

// Pass 3: out[dst] += alpha * h[src]; one wave per edge.
// Each lane owns F/32 contiguous channels, gathered with b128 loads.
__global__ void edge_agg(const int* __restrict__ src, const int* __restrict__ dst,
                         const float* __restrict__ exb, const float* __restrict__ denom,
                         const float* __restrict__ H, float* __restrict__ Out,
                         int E, int F) {
  int wave = (blockIdx.x * blockDim.x + threadIdx.x) >> 5;
  int lane = threadIdx.x & 31;
  if (wave >= E) return;
  int s = src[wave], d = dst[wave];
  float alpha = exb[wave] / fmaxf(denom[d], 1e-16f);
  int ch = F >> 5;                      // 4 (F=128) or 8 (F=256) channels/lane
  int base = lane * ch;                 // contiguous -> coalesced b128 gathers
  const float4* hv = (const float4*)(H + (size_t)s * F + base);
  float* od = Out + (size_t)d * F + base;
  for (int q = 0; q < (ch >> 2); ++q) {
    float4 v = hv[q];
    atomAddF(&od[q * 4 + 0], alpha * v.x);
    atomAddF(&od[q * 4 + 1], alpha * v.y);
    atomAddF(&od[q * 4 + 2], alpha * v.z);
    atomAddF(&od[q * 4 + 3], alpha * v.w);
  }
}

// out[i] = (relu ? max(0, io[i]+b[c]) : io[i]+b[c]); F is a power of two.
__global__ void bias_act(float* __restrict__ io, const float* __restrict__ b,
                         int total, int F, int relu) {
  int i = blockIdx.x * blockDim.x + threadIdx.x;
  if (i >= total) return;
  float v = io[i] + b[i & (F - 1)];
  if (relu) v = fmaxf(v, 0.f);
  io[i] = v;
}

// ---------- launch ----------

static inline int zgrid(long n) {
  long b = (n + 255) / 256;
  return (int)(b > 8192 ? 8192 : b);
}

extern "C" void kernel_launch(void* const* d_in, const int* in_sizes, int n_in,
                              void* d_out, int out_size, void* d_ws, size_t ws_size,
                              hipStream_t stream) {
  (void)n_in; (void)out_size;
  const float* x   = (const float*)d_in[0];
  const int*   ei  = (const int*)d_in[1];
  const float* W1  = (const float*)d_in[2];
  const float* as1 = (const float*)d_in[3];
  const float* ad1 = (const float*)d_in[4];
  const float* b1  = (const float*)d_in[5];
  const float* W2  = (const float*)d_in[6];
  const float* as2 = (const float*)d_in[7];
  const float* ad2 = (const float*)d_in[8];
  const float* b2  = (const float*)d_in[9];

  const int HID = in_sizes[3];        // 256
  const int IN  = in_sizes[2] / HID;  // 128
  const int Nn  = in_sizes[0] / IN;   // 100000
  const int OUT = in_sizes[7];        // 128
  const int E   = in_sizes[1] / 2;    // 1,700,000 (incl. self loops)
  const int* srcv = ei;
  const int* dstv = ei + E;

  // workspace carve (floats)
  float*    hbuf   = (float*)d_ws;                     // Nn*HID (h1, reused as h2)
  float*    aggbuf = hbuf + (size_t)Nn * HID;          // Nn*HID (agg1 -> relu -> layer2 in)
  float*    asv    = aggbuf + (size_t)Nn * HID;        // Nn
  float*    adv    = asv + Nn;                         // Nn
  unsigned* emax   = (unsigned*)(adv + Nn);            // Nn
  float*    denom  = (float*)(emax + Nn);              // Nn
  float*    exb    = denom + Nn;                       // E
  size_t need = ((size_t)2 * Nn * HID + 5 * (size_t)Nn + (size_t)E) * sizeof(float);
  if (ws_size < need) return;

  float* zout = (float*)d_out;

  // ---- layer 1 ----
  long z1 = (long)Nn * HID + 4L * Nn;  // agg1 + asv + adv + emax + denom
  k_zero<<<zgrid(z1), 256, 0, stream>>>(aggbuf, z1);

  {
    int tiles = ((Nn + 15) / 16) * (HID / 64);
    int blocks = (tiles + 3) / 4;  // 4 waves / 128-thread block
    gemm_wmma<<<blocks, 128, 0, stream>>>(x, W1, hbuf, Nn, IN, HID);
  }
  alpha_dots<<<(Nn + 7) / 8, 256, 0, stream>>>(hbuf, as1, ad1, asv, adv, Nn, HID);
  edge_max<<<(E + 255) / 256, 256, 0, stream>>>(srcv, dstv, asv, adv, emax, E);
  edge_exp<<<(E + 255) / 256, 256, 0, stream>>>(srcv, dstv, asv, adv, emax, denom, exb, E);
  edge_agg<<<(E + 7) / 8, 256, 0, stream>>>(srcv, dstv, exb, denom, hbuf, aggbuf, E, HID);
  bias_act<<<((Nn * HID) + 255) / 256, 256, 0, stream>>>(aggbuf, b1, Nn * HID, HID, 1);

  // ---- layer 2 ----
  k_zero<<<zgrid((long)Nn * OUT), 256, 0, stream>>>(zout, (long)Nn * OUT);
  k_zero<<<zgrid(2L * Nn), 256, 0, stream>>>((float*)emax, 2L * Nn);  // emax + denom

  {
    int tiles = ((Nn + 15) / 16) * (OUT / 64);
    int blocks = (tiles + 3) / 4;
    gemm_wmma<<<blocks, 128, 0, stream>>>(aggbuf, W2, hbuf, Nn, HID, OUT);
  }
  alpha_dots<<<(Nn + 7) / 8, 256, 0, stream>>>(hbuf, as2, ad2, asv, adv, Nn, OUT);
  edge_max<<<(E + 255) / 256, 256, 0, stream>>>(srcv, dstv, asv, adv, emax, E);
  edge_exp<<<(E + 255) / 256, 256, 0, stream>>>(srcv, dstv, asv, adv, emax, denom, exb, E);
  edge_agg<<<(E + 7) / 8, 256, 0, stream>>>(srcv, dstv, exb, denom, hbuf, zout, E, OUT);
  bias_act<<<((Nn * OUT) + 255) / 256, 256, 0, stream>>>(zout, b2, Nn * OUT, OUT, 0);
}